// AttentionGate3D_89678917140869
// MI455X (gfx1250) — compile-verified
//
#include <hip/hip_runtime.h>
#include <hip/hip_bf16.h>
#include <stdint.h>

typedef float v2f __attribute__((ext_vector_type(2)));
typedef float v8f __attribute__((ext_vector_type(8)));

#define NB    2          // batch
#define CI    128        // intermediate channels I
#define CX    128        // x channels
#define CG    256        // g channels
#define S_TOT 131072     // 32*64*64 spatial (2^17)
#define SBLK  128
#define KBLK  16
#define ASTR  18         // A LDS row stride (floats): conflict-free, 8B-aligned pairs
#define BSTR  20         // B LDS col stride (floats): 16 K + 4 pad, conflict-free
#define EPSV  1e-5f

// ---- CDNA5 async global->LDS copies (ASYNCcnt-tracked, no VGPR round trip) ----
__device__ __forceinline__ void async_lds_b32(uint32_t lds_off, const float* g) {
    uint64_t ga = (uint64_t)(uintptr_t)g;
    asm volatile("global_load_async_to_lds_b32 %0, %1, off"
                 :: "v"(lds_off), "v"(ga) : "memory");
}
__device__ __forceinline__ void async_lds_b64(uint32_t lds_off, const float* g) {
    uint64_t ga = (uint64_t)(uintptr_t)g;
    asm volatile("global_load_async_to_lds_b64 %0, %1, off"
                 :: "v"(lds_off), "v"(ga) : "memory");
}
__device__ __forceinline__ void wait_async0() {
    asm volatile("s_wait_asynccnt 0x0" ::: "memory");
}
__device__ __forceinline__ uint32_t lds_off_u32(const void* p) {
    // LDS aperture: generic addr[31:0] == LDS byte offset (ISA 10.2)
    return (uint32_t)(uintptr_t)p;
}

// ---------------------------------------------------------------------------
// C[n][128][S] = A[128][K] * B[n][K][S]   (fp32, V_WMMA_F32_16X16X4_F32)
// Block: 256 threads = 8 waves; tile 128 rows x 128 cols.
// Wave (mw = w&3, nw = w>>2): rows [32mw,32mw+32), cols [64nw,64nw+64)
// -> 2 M-tiles x 4 N-tiles per wave (8 v8f accumulators).
// Double-buffered LDS filled by GLOBAL_LOAD_ASYNC_TO_LDS; one barrier / K-block.
// A staged row-major (k-pairs contiguous), B staged col-major (k-pairs
// contiguous) so every WMMA operand loads as one ds_load_*_b64 pair (no
// register-packing moves).
// ---------------------------------------------------------------------------
__global__ __launch_bounds__(256)
void gemm128_wmma(const float* __restrict__ A, const float* __restrict__ B,
                  float* __restrict__ C, int K, long batchStrideB) {
    __shared__ float As[2][128 * ASTR];   // [r][k], stride 18
    __shared__ float Bs[2][SBLK * BSTR];  // [c][k], stride 20 (col-major)

    const int tid  = threadIdx.x;
    const int wave = tid >> 5;
    const int lane = tid & 31;
    const int half = lane >> 4;
    const int l16  = lane & 15;
    const int mw   = wave & 3;
    const int nw   = wave >> 2;

    const int  n     = blockIdx.y;
    const long sBase = (long)blockIdx.x * SBLK;
    const float* Bn  = B + (long)n * batchStrideB;
    float*       Cn  = C + (long)n * CI * (long)S_TOT;

    // A staging: 128x16 floats, 8/thread along k (4x async b64)
    const int ar  = tid >> 1;            // 0..127
    const int ak0 = (tid & 1) * 8;       // 0 or 8
    // B staging: 16x128 floats, 8/thread along k, column fixed (8x async b32)
    //   -> each async instruction reads 128 contiguous global floats across lanes
    const int bc  = tid & 127;           // column 0..127
    const int bk0 = (tid >> 7) * 8;      // 0 or 8

    auto stage = [&](int buf, int kb) {
        const float* aSrc = A + (long)ar * K + kb + ak0;
        uint32_t aOff = lds_off_u32(&As[buf][ar * ASTR + ak0]);
        async_lds_b64(aOff,      aSrc);
        async_lds_b64(aOff + 8,  aSrc + 2);
        async_lds_b64(aOff + 16, aSrc + 4);
        async_lds_b64(aOff + 24, aSrc + 6);
        const float* bSrc = Bn + (long)(kb + bk0) * S_TOT + sBase + bc;
        uint32_t bOff = lds_off_u32(&Bs[buf][bc * BSTR + bk0]);
#pragma unroll
        for (int j = 0; j < 8; ++j)
            async_lds_b32(bOff + 4 * j, bSrc + (long)j * S_TOT);
    };

    const v8f zero = {0.f,0.f,0.f,0.f,0.f,0.f,0.f,0.f};
    v8f acc[2][4];
#pragma unroll
    for (int mt = 0; mt < 2; ++mt)
#pragma unroll
        for (int nt = 0; nt < 4; ++nt) acc[mt][nt] = zero;

    stage(0, 0);
    const int nkb = K / KBLK;

    for (int i = 0; i < nkb; ++i) {
        wait_async0();        // own async writes to buf[i&1] complete
        __syncthreads();      // all waves' writes visible; all done reading buf[(i+1)&1]
        if (i + 1 < nkb) stage((i + 1) & 1, (i + 1) * KBLK);

        const float* Al = As[i & 1];
        const float* Bl = Bs[i & 1];
#pragma unroll
        for (int ks = 0; ks < KBLK; ks += 4) {
            v2f a[2];
#pragma unroll
            for (int mt = 0; mt < 2; ++mt) {
                const float* ap = &Al[(32*mw + 16*mt + l16) * ASTR + ks + 2*half];
                a[mt].x = ap[0];
                a[mt].y = ap[1];
            }
#pragma unroll
            for (int nt = 0; nt < 4; ++nt) {
                const float* bp = &Bl[(64*nw + 16*nt + l16) * BSTR + ks + 2*half];
                v2f b;
                b.x = bp[0];
                b.y = bp[1];
#pragma unroll
                for (int mt = 0; mt < 2; ++mt) {
                    acc[mt][nt] = __builtin_amdgcn_wmma_f32_16x16x4_f32(
                        false, a[mt], false, b, (short)0, acc[mt][nt],
                        false, false);
                }
            }
        }
    }

    // store: C/D layout: VGPR p -> M = p + 8*half, N = l16
#pragma unroll
    for (int mt = 0; mt < 2; ++mt)
#pragma unroll
        for (int nt = 0; nt < 4; ++nt)
#pragma unroll
            for (int p = 0; p < 8; ++p) {
                const int row = 32*mw + 16*mt + p + 8*half;
                Cn[(long)row * S_TOT + sBase + 64*nw + 16*nt + l16] = acc[mt][nt][p];
            }
}

// ---------------------------------------------------------------------------
// Per-row mean / rsqrt(var+eps) over S. One block per row. Deterministic.
// ---------------------------------------------------------------------------
__global__ __launch_bounds__(256)
void row_stats(const float* __restrict__ buf, float* __restrict__ out) {
    __shared__ float ssum[256];
    __shared__ float ssq[256];
    const int row = blockIdx.x;
    const float4* p4 = (const float4*)(buf + (long)row * S_TOT);
    float s0 = 0.f, s1 = 0.f;
    for (int i = threadIdx.x; i < S_TOT / 4; i += 256) {
        float4 v = p4[i];
        s0 += v.x + v.y + v.z + v.w;
        s1 += v.x*v.x + v.y*v.y + v.z*v.z + v.w*v.w;
    }
    ssum[threadIdx.x] = s0;
    ssq[threadIdx.x]  = s1;
    __syncthreads();
    for (int off = 128; off > 0; off >>= 1) {
        if (threadIdx.x < off) {
            ssum[threadIdx.x] += ssum[threadIdx.x + off];
            ssq[threadIdx.x]  += ssq[threadIdx.x + off];
        }
        __syncthreads();
    }
    if (threadIdx.x == 0) {
        const float mean = ssum[0] * (1.0f / S_TOT);
        const float var  = ssq[0] * (1.0f / S_TOT) - mean * mean;
        out[row * 2 + 0] = mean;
        out[row * 2 + 1] = rsqrtf(var + EPSV);
    }
}

// ---------------------------------------------------------------------------
// psi[n][s] = sum_i Wpsi[i] * relu( IN(u)[n,i,s] + IN(v)[n,i,s] )
// statsUV rows: 0..255 = u (n*128+i), 256..511 = v
// ---------------------------------------------------------------------------
__global__ __launch_bounds__(256)
void psi_kernel(const float* __restrict__ u, const float* __restrict__ v,
                const float* __restrict__ statsUV,
                const float* __restrict__ Wpsi,
                const float* __restrict__ gamma_g, const float* __restrict__ beta_g,
                const float* __restrict__ gamma_x, const float* __restrict__ beta_x,
                float* __restrict__ psi) {
    const int  n = blockIdx.y;
    const long s = (long)blockIdx.x * blockDim.x + threadIdx.x;
    float acc = 0.f;
#pragma unroll 4
    for (int i = 0; i < CI; ++i) {
        const int r = n * CI + i;
        const float mu_g = statsUV[r * 2 + 0];
        const float rs_g = statsUV[r * 2 + 1];
        const float mu_x = statsUV[(256 + r) * 2 + 0];
        const float rs_x = statsUV[(256 + r) * 2 + 1];
        const float uu = u[(long)r * S_TOT + s];
        const float vv = v[(long)r * S_TOT + s];
        const float hg = (uu - mu_g) * rs_g * gamma_g[i] + beta_g[i];
        const float hx = (vv - mu_x) * rs_x * gamma_x[i] + beta_x[i];
        const float h  = fmaxf(hg + hx, 0.f);
        acc = fmaf(Wpsi[i], h, acc);
    }
    psi[(long)n * S_TOT + s] = acc;
}

// ---------------------------------------------------------------------------
// out = x * sigmoid( (psi - mu)*rstd*gamma_p + beta_p ), float4-vectorized
// ---------------------------------------------------------------------------
__global__ __launch_bounds__(256)
void finalize4(const float4* __restrict__ x, const float* __restrict__ psi,
               const float* __restrict__ statsP,
               const float* __restrict__ gamma_p, const float* __restrict__ beta_p,
               float4* __restrict__ out, long total4) {
    const long idx = (long)blockIdx.x * blockDim.x + threadIdx.x;
    if (idx >= total4) return;
    const long e = idx * 4;
    const long s = e & (long)(S_TOT - 1);      // S is 2^17
    const long n = e >> 24;                    // e / (S * 128)
    const float mu = statsP[n * 2 + 0];
    const float rs = statsP[n * 2 + 1];
    const float gp = gamma_p[0];
    const float bp = beta_p[0];
    const float4 pv = *(const float4*)(psi + n * S_TOT + s);
    const float4 xv = x[idx];
    float4 o;
    {
        float t;
        t = (pv.x - mu) * rs * gp + bp; o.x = xv.x / (1.f + __expf(-t));
        t = (pv.y - mu) * rs * gp + bp; o.y = xv.y / (1.f + __expf(-t));
        t = (pv.z - mu) * rs * gp + bp; o.z = xv.z / (1.f + __expf(-t));
        t = (pv.w - mu) * rs * gp + bp; o.w = xv.w / (1.f + __expf(-t));
    }
    out[idx] = o;
}

// ---------------------------------------------------------------------------
extern "C" void kernel_launch(void* const* d_in, const int* in_sizes, int n_in,
                              void* d_out, int out_size, void* d_ws, size_t ws_size,
                              hipStream_t stream) {
    const float* g       = (const float*)d_in[0];   // [2,256,S]
    const float* x       = (const float*)d_in[1];   // [2,128,S]
    const float* Wg      = (const float*)d_in[2];   // [128,256]
    const float* Wx      = (const float*)d_in[3];   // [128,128]
    const float* Wpsi    = (const float*)d_in[4];   // [1,128]
    const float* gamma_g = (const float*)d_in[5];
    const float* beta_g  = (const float*)d_in[6];
    const float* gamma_x = (const float*)d_in[7];
    const float* beta_x  = (const float*)d_in[8];
    const float* gamma_p = (const float*)d_in[9];
    const float* beta_p  = (const float*)d_in[10];
    float* out = (float*)d_out;

    float* ws = (float*)d_ws;
    const long U_ELEMS = (long)NB * CI * S_TOT;     // 33,554,432
    float* u       = ws;                            // [2,128,S]
    float* v       = ws + U_ELEMS;                  // [2,128,S] (contiguous after u)
    float* psi     = ws + 2 * U_ELEMS;              // [2,S]
    float* statsUV = ws + 2 * U_ELEMS + (long)NB * S_TOT;   // 512 rows * 2
    float* statsP  = statsUV + 1024;                        // 2 rows * 2

    dim3 blk(256);

    // GEMM 1: u = Wg @ g  (K = 256)
    gemm128_wmma<<<dim3(S_TOT / SBLK, NB), blk, 0, stream>>>(
        Wg, g, u, CG, (long)CG * S_TOT);
    // GEMM 2: v = Wx @ x  (K = 128)
    gemm128_wmma<<<dim3(S_TOT / SBLK, NB), blk, 0, stream>>>(
        Wx, x, v, CX, (long)CX * S_TOT);

    // Instance-norm stats for u (rows 0..255) and v (rows 256..511)
    row_stats<<<dim3(512), blk, 0, stream>>>(u, statsUV);

    // psi = Wpsi . relu(IN(u)+IN(v))
    psi_kernel<<<dim3(S_TOT / 256, NB), blk, 0, stream>>>(
        u, v, statsUV, Wpsi, gamma_g, beta_g, gamma_x, beta_x, psi);

    // psi instance-norm stats (2 rows)
    row_stats<<<dim3(2), blk, 0, stream>>>(psi, statsP);

    // out = x * sigmoid(IN(psi))
    const long total4 = (long)NB * CX * S_TOT / 4;  // 8,388,608
    finalize4<<<dim3((unsigned)((total4 + 255) / 256)), blk, 0, stream>>>(
        (const float4*)x, psi, statsP, gamma_p, beta_p, (float4*)out, total4);
}